// PointConv_6047313953091
// MI455X (gfx1250) — compile-verified
//
#include <hip/hip_runtime.h>

#define RADIUS_F 4.0f
#define WPB 4   // waves per block (128 threads)

typedef float v2f __attribute__((ext_vector_type(2)));
typedef float v8f __attribute__((ext_vector_type(8)));

// ---------------------------------------------------------------------------
// ws layout (floats): [0..12) sum/sumsq accumulators, [12..18) inv_scale,
// [64 .. 64 + B*N*64) transposed features (b, N, 64)
// ---------------------------------------------------------------------------

__global__ void zero_accum_k(float* accum) {
    if (threadIdx.x < 12) accum[threadIdx.x] = 0.0f;
}

__device__ inline float wave_sum32(float v) {
    #pragma unroll
    for (int o = 16; o >= 1; o >>= 1) v += __shfl_xor(v, o, 32);
    return v;
}

// Partial sums of rel and rel^2 per (b, coord-channel); blockIdx.y = batch
__global__ void std_partial_k(const float* __restrict__ xyz,
                              const int* __restrict__ knn,
                              float* __restrict__ accum, int N) {
    const int b = blockIdx.y;
    const float* xb = xyz + (size_t)b * 3 * N;
    float s0 = 0.f, s1 = 0.f, s2 = 0.f, q0 = 0.f, q1 = 0.f, q2 = 0.f;
    for (size_t n = (size_t)blockIdx.x * blockDim.x + threadIdx.x; n < (size_t)N;
         n += (size_t)gridDim.x * blockDim.x) {
        const int* kp = knn + ((size_t)b * N + n) * 16;
        float c0 = xb[n], c1 = xb[(size_t)N + n], c2 = xb[2 * (size_t)N + n];
        #pragma unroll 4
        for (int j = 0; j < 16; ++j) {
            int i = kp[j];
            float r0 = xb[i] - c0;
            float r1 = xb[(size_t)N + i] - c1;
            float r2 = xb[2 * (size_t)N + i] - c2;
            s0 += r0; q0 += r0 * r0;
            s1 += r1; q1 += r1 * r1;
            s2 += r2; q2 += r2 * r2;
        }
    }
    s0 = wave_sum32(s0); q0 = wave_sum32(q0);
    s1 = wave_sum32(s1); q1 = wave_sum32(q1);
    s2 = wave_sum32(s2); q2 = wave_sum32(q2);
    if ((threadIdx.x & 31) == 0) {
        atomicAdd(&accum[b * 3 + 0], s0); atomicAdd(&accum[6 + b * 3 + 0], q0);
        atomicAdd(&accum[b * 3 + 1], s1); atomicAdd(&accum[6 + b * 3 + 1], q1);
        atomicAdd(&accum[b * 3 + 2], s2); atomicAdd(&accum[6 + b * 3 + 2], q2);
    }
}

__global__ void finalize_k(const float* __restrict__ accum,
                           float* __restrict__ inv, int N) {
    int i = threadIdx.x;
    if (i < 6) {
        float M = 16.0f * (float)N;
        float s = accum[i], q = accum[6 + i];
        float var = (q - s * (s / M)) / (M - 1.0f);       // ddof = 1
        inv[i] = 1.0f / (RADIUS_F * sqrtf(fmaxf(var, 1e-30f)));
    }
}

// fea (b, 64, N) -> fea_t (b, N, 64) so neighbor gathers are contiguous 256B
__global__ void transpose_k(const float* __restrict__ fea,
                            float* __restrict__ fea_t, int B, int N) {
    size_t total = (size_t)B * 64 * N;
    for (size_t t = (size_t)blockIdx.x * blockDim.x + threadIdx.x; t < total;
         t += (size_t)gridDim.x * blockDim.x) {
        size_t bn = t / ((size_t)64 * N);
        size_t rem = t - bn * (size_t)64 * N;
        size_t c = rem / N, n = rem - c * N;
        fea_t[(bn * N + n) * 64 + c] = fea[t];
    }
}

// ---------------------------------------------------------------------------
// Main fused kernel: one wave per point, double-buffered async gather.
// Raw rows Y = W*[center, nb_1..nb_15] + bias computed with fp32 WMMA;
// center-diff applied post-GEMM via Y_j - Y_0 + bias (linearity).
// ---------------------------------------------------------------------------
__global__ __launch_bounds__(128) void pointconv_main_k(
    const float* __restrict__ xyz,
    const float* __restrict__ fea_t,
    const int* __restrict__ knn,
    const float* __restrict__ conv_w,
    const float* __restrict__ conv_b,
    const float* __restrict__ conv_dw,   // (64,125) flattened
    const float* __restrict__ inv_scale, // 6 floats
    float* __restrict__ out,             // (B,128,N)
    int B, int N) {
    // W fragments: [mt*16+kk][lane][v] -> 4*16*64 floats
    __shared__ float lds_W[4096];
    // Raw gathered rows, double buffered: [wave][buf][j*68 + c] (68 = padded)
    __shared__ float lds_X[WPB][2][16 * 68];
    __shared__ int   lds_cell[WPB][16];
    __shared__ float lds_inv[6];

    const int tid  = threadIdx.x;
    const int wave = tid >> 5;
    const int lane = tid & 31;
    const int hi   = lane >> 4;   // half-wave
    const int lm   = lane & 15;   // N-column / M-row within tile

    // Stage W into A-fragment order: lane L holds W[m=mt*16+lm][k=4kk+2*hi+v]
    for (int f = tid; f < 4096; f += 128) {
        int mt  = f >> 10;
        int rem = f & 1023;
        int kk  = rem >> 6;
        int L   = (rem >> 1) & 31;
        int v   = rem & 1;
        int o   = mt * 16 + (L & 15);
        int c   = kk * 4 + ((L >> 4) << 1) + v;
        lds_W[f] = conv_w[o * 64 + c];
    }
    if (tid < 6) lds_inv[tid] = inv_scale[tid];
    __syncthreads();

    // Bias in registers: bias[mt][r] feeds C/D slot (M = mt*16 + hi*8 + r)
    float bias[4][8];
    #pragma unroll
    for (int mt = 0; mt < 4; ++mt)
        #pragma unroll
        for (int r = 0; r < 8; ++r)
            bias[mt][r] = conv_b[mt * 16 + hi * 8 + r];

    const int jg = lane >> 1;   // column this lane gathers
    const int h  = lane & 1;    // channel half (32 floats = 128 B)

    // Issue 8 async b128 loads: row jg (center for jg==0), channels h*32..+31
    auto issue = [&](size_t p, int bufi) {
        int b = (int)(p / (size_t)N);
        int n = (int)(p - (size_t)b * N);
        const int* kp = knn + p * 16;
        int src = (jg == 0) ? n : kp[jg];
        const float* gsrc = fea_t + ((size_t)b * N + src) * 64 + h * 32;
        unsigned la = (unsigned)(size_t)&lds_X[wave][bufi][jg * 68 + h * 32];
        unsigned long long ga = (unsigned long long)(size_t)gsrc;
        asm volatile(
            "global_load_async_to_lds_b128 %0, %1, off\n\t"
            "global_load_async_to_lds_b128 %0, %1, off offset:16\n\t"
            "global_load_async_to_lds_b128 %0, %1, off offset:32\n\t"
            "global_load_async_to_lds_b128 %0, %1, off offset:48\n\t"
            "global_load_async_to_lds_b128 %0, %1, off offset:64\n\t"
            "global_load_async_to_lds_b128 %0, %1, off offset:80\n\t"
            "global_load_async_to_lds_b128 %0, %1, off offset:96\n\t"
            "global_load_async_to_lds_b128 %0, %1, off offset:112"
            :: "v"(la), "v"(ga) : "memory");
    };

    const size_t total = (size_t)B * N;
    const size_t gw = (size_t)blockIdx.x * WPB + wave;
    const size_t nw = (size_t)gridDim.x * WPB;

    int bufi = 0;
    if (gw < total) issue(gw, 0);

    for (size_t p = gw; p < total; p += nw) {
        const int b = (int)(p / (size_t)N);
        const int n = (int)(p - (size_t)b * N);
        const int* kp = knn + p * 16;

        // Prefetch next point's rows into the other buffer
        const size_t pn = p + nw;
        const bool have_next = pn < total;
        if (have_next) issue(pn, bufi ^ 1);

        // ---- Phase 1: 5^3 cell index per neighbor (lanes 0..15) ----
        if (lane < 16) {
            int idxj = kp[lane];
            int cell = 0, w = 1;
            #pragma unroll
            for (int c = 0; c < 3; ++c) {
                float ctr = xyz[((size_t)b * 3 + c) * N + n];
                float nb  = xyz[((size_t)b * 3 + c) * N + idxj];
                float sc  = (nb - ctr) * lds_inv[b * 3 + c];
                // ((sc+1)*5-1)*0.5 = 2.5*sc + 2; round-half-even then clip
                float fx = rintf(2.5f * sc + 2.0f);
                fx = fminf(fmaxf(fx, 0.0f), 4.0f);
                cell += (int)fx * w;
                w *= 5;
            }
            lds_cell[wave][lane] = cell;
        }

        // Current buffer's 8 async loads are the oldest outstanding group;
        // async loads complete in order, so <=8 retires them.
        if (have_next) asm volatile("s_wait_asynccnt 0x8" ::: "memory");
        else           asm volatile("s_wait_asynccnt 0x0" ::: "memory");
        __builtin_amdgcn_wave_barrier();

        // ---- Phase 3: fp32 WMMA GEMM on raw rows ----
        const float* Xb = lds_X[wave][bufi];
        v8f a0, a1, a2, a3;
        #pragma unroll
        for (int r = 0; r < 8; ++r) {
            a0[r] = bias[0][r]; a1[r] = bias[1][r];
            a2[r] = bias[2][r]; a3[r] = bias[3][r];
        }
        #pragma unroll
        for (int kk = 0; kk < 16; ++kk) {
            // B fragment: lane holds X[c = 4kk+2hi+{0,1}][j = lm]
            v2f bf = *(const v2f*)&Xb[lm * 68 + 4 * kk + 2 * hi];
            v2f w0 = ((v2f*)&lds_W[(0 * 16 + kk) * 64])[lane];
            v2f w1 = ((v2f*)&lds_W[(1 * 16 + kk) * 64])[lane];
            v2f w2 = ((v2f*)&lds_W[(2 * 16 + kk) * 64])[lane];
            v2f w3 = ((v2f*)&lds_W[(3 * 16 + kk) * 64])[lane];
            a0 = __builtin_amdgcn_wmma_f32_16x16x4_f32(false, w0, false, bf,
                                                       (short)0, a0, false, false);
            a1 = __builtin_amdgcn_wmma_f32_16x16x4_f32(false, w1, false, bf,
                                                       (short)0, a1, false, false);
            a2 = __builtin_amdgcn_wmma_f32_16x16x4_f32(false, w2, false, bf,
                                                       (short)0, a2, false, false);
            a3 = __builtin_amdgcn_wmma_f32_16x16x4_f32(false, w3, false, bf,
                                                       (short)0, a3, false, false);
        }

        // ---- Phase 4: center-diff fixup, depthwise weight, reduce, store ----
        int cell_j = lds_cell[wave][lm];
        #pragma unroll
        for (int mt = 0; mt < 4; ++mt) {
            v8f acc = (mt == 0) ? a0 : (mt == 1) ? a1 : (mt == 2) ? a2 : a3;
            #pragma unroll
            for (int r = 0; r < 8; ++r) {
                int o = mt * 16 + hi * 8 + r;
                float y  = acc[r];                 // W*nb_j + bias (col j = lm)
                float y0 = __shfl(y, 0, 16);       // column 0 = W*center + bias
                float nf = (lm == 0) ? y : (y - y0 + bias[mt][r]);
                float s1 = conv_dw[o * 125 + cell_j] * nf;
                float s2 = nf;
                #pragma unroll
                for (int off = 8; off >= 1; off >>= 1) {
                    s1 += __shfl_xor(s1, off, 16);
                    s2 += __shfl_xor(s2, off, 16);
                }
                if (lm == 0) {
                    size_t ob = ((size_t)b * 128 + o) * N + n;
                    out[ob] = s1;                              // conv half
                    out[ob + (size_t)64 * N] = s2 * 0.0625f;   // mean half
                }
            }
        }
        bufi ^= 1;
    }
}

// ---------------------------------------------------------------------------
extern "C" void kernel_launch(void* const* d_in, const int* in_sizes, int n_in,
                              void* d_out, int out_size, void* d_ws, size_t ws_size,
                              hipStream_t stream) {
    const float* xyz     = (const float*)d_in[0];
    const float* fea     = (const float*)d_in[1];
    const int*   knn     = (const int*)d_in[2];
    const float* conv_w  = (const float*)d_in[3];
    const float* conv_b  = (const float*)d_in[4];
    const float* conv_dw = (const float*)d_in[5];
    float* out = (float*)d_out;
    float* ws  = (float*)d_ws;

    const int B = 2;                       // per reference setup
    const int N = in_sizes[0] / (3 * B);   // xyz is (B,3,N)

    float* accum = ws;        // 12 floats
    float* inv   = ws + 12;   // 6 floats
    float* fea_t = ws + 64;   // B*N*64 floats

    hipLaunchKernelGGL(zero_accum_k, dim3(1), dim3(32), 0, stream, accum);
    hipLaunchKernelGGL(std_partial_k, dim3(256, B), dim3(256), 0, stream,
                       xyz, knn, accum, N);
    hipLaunchKernelGGL(finalize_k, dim3(1), dim3(32), 0, stream, accum, inv, N);
    hipLaunchKernelGGL(transpose_k, dim3(2048), dim3(256), 0, stream,
                       fea, fea_t, B, N);
    hipLaunchKernelGGL(pointconv_main_k, dim3(2048), dim3(128), 0, stream,
                       xyz, fea_t, knn, conv_w, conv_b, conv_dw, inv, out, B, N);
}